// SAGEConvAgg_88734024335500
// MI455X (gfx1250) — compile-verified
//
#include <hip/hip_runtime.h>

// ---------------------------------------------------------------------------
// SAGE mean aggregation: neigh[v] = mean_{(u->v)} feat[u]   (N=100k, E=1M, D=64)
// Memory-bound (whole working set ~60MB fits MI455X's 192MB L2). Limiter is
// L2 fp32-atomic throughput, so: non-returning global_atomic_add_f32,
// B128 gathers of the L2-resident feature rows, wave32 16-lane-per-edge
// mapping, and CDNA5 async-to-LDS staging of the edge-index stream.
// NOTE: scatter kernel is deliberately first in the file so the disasm
// snippet shows its body (async-to-LDS + atomic lowering check).
// ---------------------------------------------------------------------------

#define AS1 __attribute__((address_space(1)))
#define AS3 __attribute__((address_space(3)))

__device__ __forceinline__ void async_copy_b32(const int* g, int* l) {
#if __has_builtin(__builtin_amdgcn_global_load_async_to_lds_b32)
  __builtin_amdgcn_global_load_async_to_lds_b32((AS1 int*)g, (AS3 int*)l, 0, 0);
#else
  unsigned loff = (unsigned)(unsigned long long)(AS3 int*)l;  // LDS byte offset
  asm volatile("global_load_async_to_lds_b32 %0, %1, off"
               :: "v"(loff), "v"(g) : "memory");
#endif
}

__device__ __forceinline__ void wait_async0() {
#if __has_builtin(__builtin_amdgcn_s_wait_asynccnt)
  __builtin_amdgcn_s_wait_asynccnt(0);
#else
  asm volatile("s_wait_asynccnt 0" ::: "memory");
#endif
}

__device__ __forceinline__ void atom_add_f32(float* p, float v) {
  // Relaxed, agent scope, result unused -> non-returning global_atomic_add_f32
  (void)__hip_atomic_fetch_add(p, v, __ATOMIC_RELAXED, __HIP_MEMORY_SCOPE_AGENT);
}

// ---------------- kernel: edge-parallel scatter-add (FIRST in file) ---------
// Block = 256 threads = 8 waves. Block owns a 256-edge chunk whose src/dst
// indices are async-copied to LDS. 16 lanes per edge, one float4 per lane.
#define CHUNK 256
__global__ void sage_scatter_kernel(const float* __restrict__ feat,
                                    const int* __restrict__ src,
                                    const int* __restrict__ dst,
                                    float* __restrict__ outsum,
                                    float* __restrict__ deg, int E) {
  __shared__ int s_src[CHUNK];
  __shared__ int s_dst[CHUNK];

  const int tid  = threadIdx.x;               // 0..255
  const int base = blockIdx.x * CHUNK;

  // Stage this chunk's edge indices into LDS via the async copy engine.
  int gi = base + tid;
  int ci = gi < E ? gi : (E - 1);             // clamp: no OOB global reads
  async_copy_b32(src + ci, &s_src[tid]);
  async_copy_b32(dst + ci, &s_dst[tid]);
  wait_async0();
  __syncthreads();

  const int group = tid >> 4;                 // 16 edge-groups per block
  const int lg    = tid & 15;                 // lane-in-group: one float4 each

  for (int e = group; e < CHUNK; e += 16) {
    int ge = base + e;
    if (ge >= E) break;                       // tail chunk
    int s = s_src[e];
    int d = s_dst[e];

    // 256B feature row gather: 16 lanes x global_load_b128 (L2-resident)
    const float4 v = ((const float4*)feat)[(size_t)s * 16 + lg];

    float* o = outsum + (size_t)d * 64 + lg * 4;
    atom_add_f32(o + 0, v.x);
    atom_add_f32(o + 1, v.y);
    atom_add_f32(o + 2, v.z);
    atom_add_f32(o + 3, v.w);
    if (lg == 0) atom_add_f32(&deg[d], 1.0f);
  }
}

// ---------------- kernel: zero accumulator (d_out) and degrees (d_ws) -------
__global__ void sage_zero_kernel(float* __restrict__ out,
                                 float* __restrict__ deg, int N) {
  int t = blockIdx.x * blockDim.x + threadIdx.x;
  int nquads = N * 16;               // N*64 floats as float4
  if (t < nquads) {
    float4 z = {0.f, 0.f, 0.f, 0.f};
    ((float4*)out)[t] = z;
  }
  if (t < N) deg[t] = 0.f;
}

// ---------------- kernel: normalize in place --------------------------------
__global__ void sage_normalize_kernel(float* __restrict__ out,
                                      const float* __restrict__ deg, int N) {
  int t = blockIdx.x * blockDim.x + threadIdx.x;
  int node = t >> 4;
  int q    = t & 15;
  if (node >= N) return;
  float d   = deg[node];
  float inv = (d > 0.f) ? (1.f / d) : 0.f;    // == where(deg>0, sum/deg, 0)
  float4* p = (float4*)out + (size_t)node * 16 + q;
  float4 v  = *p;
  v.x *= inv; v.y *= inv; v.z *= inv; v.w *= inv;
  *p = v;
}

// ---------------------------------------------------------------------------
extern "C" void kernel_launch(void* const* d_in, const int* in_sizes, int n_in,
                              void* d_out, int out_size, void* d_ws, size_t ws_size,
                              hipStream_t stream) {
  const float* feat = (const float*)d_in[0];
  const int*   src  = (const int*)d_in[1];
  const int*   dst  = (const int*)d_in[2];
  float*       out  = (float*)d_out;
  float*       deg  = (float*)d_ws;           // N floats of scratch

  const int N = in_sizes[0] / 64;             // D = 64
  const int E = in_sizes[1];

  const int zthreads = N * 16;
  sage_zero_kernel<<<(zthreads + 255) / 256, 256, 0, stream>>>(out, deg, N);

  const int nChunks = (E + CHUNK - 1) / CHUNK;
  sage_scatter_kernel<<<nChunks, 256, 0, stream>>>(feat, src, dst, out, deg, E);

  sage_normalize_kernel<<<(N * 16 + 255) / 256, 256, 0, stream>>>(out, deg, N);
}